// PointcloudGrouping_39960375722107
// MI455X (gfx1250) — compile-verified
//
#include <hip/hip_runtime.h>
#include <hip/hip_bf16.h>

// ---------------- problem constants ----------------
constexpr int NB   = 8;       // batches
constexpr int NPTS = 16384;   // points per batch
constexpr int NG   = 512;     // FPS groups
constexpr int KS   = 32;      // group size after energy top-k
constexpr int KUP  = 512;     // ball-query candidate cap
constexpr int CTX  = 256;     // context length
constexpr float R2 = 0.25f;   // radius^2

// output layout (floats, reference return order)
constexpr size_t OUT_GROUPS  = 0;                       // 8*256*32*4
constexpr size_t OUT_CENTERS = 262144;                  // 8*256*4
constexpr size_t OUT_EMB     = 270336;                  // 8*256
constexpr size_t OUT_PMASK   = 272384;                  // 8*256*32

typedef float v2f __attribute__((ext_vector_type(2)));
typedef float v8f __attribute__((ext_vector_type(8)));

__device__ __forceinline__ float neg_inf() { return -__builtin_inff(); }

// ============================================================
// Kernel 1: farthest point sampling, one block per batch.
// ============================================================
__global__ __launch_bounds__(1024) void fps_kernel(const float* __restrict__ pts,
                                                   const int* __restrict__ lengths,
                                                   float* __restrict__ ws_centers,
                                                   int* __restrict__ ws_glen,
                                                   float* __restrict__ out) {
  __shared__ float redV[32];
  __shared__ int   redI[32];
  __shared__ int   sel[NG];
  __shared__ float lastxyz[3];

  const int b    = blockIdx.x;
  const int t    = threadIdx.x;
  const int lane = t & 31;
  const int wv   = t >> 5;
  const int len  = lengths[b];
  const float4* pp = (const float4*)(pts + (size_t)b * NPTS * 4);

  float mind[16];
#pragma unroll
  for (int k = 0; k < 16; ++k) mind[k] = 1e10f;

  if (t == 0) {
    sel[0] = 0;
    float4 p0 = pp[0];
    lastxyz[0] = p0.x; lastxyz[1] = p0.y; lastxyz[2] = p0.z;
    ws_glen[b] = 0;                      // zero group-length accumulator
  }
  __syncthreads();

  for (int it = 1; it < NG; ++it) {
    const float lx = lastxyz[0], ly = lastxyz[1], lz = lastxyz[2];
    float bv = neg_inf();
    int   bi = 0x7fffffff;
#pragma unroll
    for (int k = 0; k < 16; ++k) {
      const int p = t + (k << 10);
      float4 q = pp[p];
      const float dx = q.x - lx, dy = q.y - ly, dz = q.z - lz;
      const float d = dx * dx + dy * dy + dz * dz;
      mind[k] = fminf(mind[k], d);
      const float key = (p < len) ? mind[k] : neg_inf();
      if (key > bv || (key == bv && p < bi)) { bv = key; bi = p; }
    }
    // wave32 argmax reduce (tie -> lower index, matching jnp.argmax)
#pragma unroll
    for (int d = 16; d >= 1; d >>= 1) {
      const float ov = __shfl_xor(bv, d);
      const int   oi = __shfl_xor(bi, d);
      if (ov > bv || (ov == bv && oi < bi)) { bv = ov; bi = oi; }
    }
    if (lane == 0) { redV[wv] = bv; redI[wv] = bi; }
    __syncthreads();
    if (wv == 0) {
      bv = redV[lane]; bi = redI[lane];
#pragma unroll
      for (int d = 16; d >= 1; d >>= 1) {
        const float ov = __shfl_xor(bv, d);
        const int   oi = __shfl_xor(bi, d);
        if (ov > bv || (ov == bv && oi < bi)) { bv = ov; bi = oi; }
      }
      if (lane == 0) {
        sel[it] = bi;
        float4 q = pp[bi];
        lastxyz[0] = q.x; lastxyz[1] = q.y; lastxyz[2] = q.z;
      }
    }
    __syncthreads();
  }

  // gather centers (all 4 channels, like masked_gather(points, idx))
  if (t < NG) {
    float4 q = pp[sel[t]];
    ((float4*)ws_centers)[(size_t)b * NG + t] = q;
    if (t < CTX) ((float4*)(out + OUT_CENTERS))[(size_t)b * CTX + t] = q;
  }
}

// ============================================================
// Kernel 2: WMMA ball-query bitmask.
// Block = 8 waves, one 16-center tile per block; each wave sweeps
// 16-point chunks. One V_WMMA_F32_16X16X4_F32 per chunk computes
// c·(-2p)+|p|^2 for 16 centers x 16 points; ballots turn the 16x16
// predicate tile into 16-bit masks (bit j = point j of the chunk).
// ============================================================
__global__ __launch_bounds__(256) void ball_mask_kernel(const float* __restrict__ pts,
                                                        const float* __restrict__ centers,
                                                        const int* __restrict__ lengths,
                                                        unsigned short* __restrict__ mask) {
  const int tile = blockIdx.x;          // 0..255
  const int b    = tile >> 5;           // 32 tiles per batch
  const int g0   = (tile & 31) << 4;    // first center of tile
  const int wv   = threadIdx.x >> 5;
  const int lane = threadIdx.x & 31;
  const int m    = lane & 15;           // column / center-row select
  const int h    = lane >> 4;           // K-half
  const int len  = lengths[b];

  // A operand: row m = (cx, cy, cz, 1); lanes<16 hold K0/K1, lanes>=16 K2/K3
  float4 cm = ((const float4*)(centers))[(size_t)b * NG + g0 + m];
  v2f a;
  a[0] = h ? cm.z : cm.x;
  a[1] = h ? 1.0f : cm.y;
  const float mynorm = cm.x * cm.x + cm.y * cm.y + cm.z * cm.z;
  // per-row threshold: D < 0.25 - |c_row|^2 ; D rows are r (lanes<16) / r+8 (lanes>=16)
  float thr[8];
#pragma unroll
  for (int r = 0; r < 8; ++r) thr[r] = R2 - __shfl(mynorm, r + 8 * h);

  const float4* pp = (const float4*)(pts + (size_t)b * NPTS * 4);
  unsigned short* mrow = mask + ((size_t)b * NG + g0) * (NPTS / 16);

  for (int c = wv; c < NPTS / 16; c += 8) {
    const int n0 = c << 4;
    float4 p = pp[n0 + m];
    const float pn = p.x * p.x + p.y * p.y + p.z * p.z;
    v2f bb;
    bb[0] = h ? (-2.0f * p.z) : (-2.0f * p.x);
    bb[1] = h ? pn            : (-2.0f * p.y);
    v8f acc = {};
    acc = __builtin_amdgcn_wmma_f32_16x16x4_f32(false, a, false, bb,
                                                (short)0, acc, false, false);
    const bool valid = (n0 + m) < len;
    const unsigned b0 = (unsigned)__ballot(valid && (acc[0] < thr[0]));
    const unsigned b1 = (unsigned)__ballot(valid && (acc[1] < thr[1]));
    const unsigned b2 = (unsigned)__ballot(valid && (acc[2] < thr[2]));
    const unsigned b3 = (unsigned)__ballot(valid && (acc[3] < thr[3]));
    const unsigned b4 = (unsigned)__ballot(valid && (acc[4] < thr[4]));
    const unsigned b5 = (unsigned)__ballot(valid && (acc[5] < thr[5]));
    const unsigned b6 = (unsigned)__ballot(valid && (acc[6] < thr[6]));
    const unsigned b7 = (unsigned)__ballot(valid && (acc[7] < thr[7]));
    if (lane < 16) {
      const unsigned rr = lane & 7u;      // ballot index; low16 = row rr, high16 = row rr+8
      unsigned v = b0;
      v = (rr == 1) ? b1 : v; v = (rr == 2) ? b2 : v; v = (rr == 3) ? b3 : v;
      v = (rr == 4) ? b4 : v; v = (rr == 5) ? b5 : v; v = (rr == 6) ? b6 : v;
      v = (rr == 7) ? b7 : v;
      const unsigned m16 = (lane < 8) ? (v & 0xffffu) : (v >> 16);
      mrow[(size_t)lane * (NPTS / 16) + c] = (unsigned short)m16;
    }
  }
}

// ============================================================
// Kernel 3: per-group ordered collection + energy top-32.
// One wave32 per group; 4 waves per block.
// ============================================================
__global__ __launch_bounds__(128) void group_kernel(const float* __restrict__ pts,
                                                    const unsigned* __restrict__ maskWords,
                                                    int* __restrict__ ws_glen,
                                                    float* __restrict__ out) {
  __shared__ float candE[4][KUP];
  __shared__ int   candI[4][KUP];

  const int lane = threadIdx.x & 31;
  const int w    = threadIdx.x >> 5;
  const int gid  = blockIdx.x * 4 + w;      // 0..4095
  const int b    = gid >> 9;
  const int gi   = gid & 511;
  const unsigned* mw = maskWords + (size_t)gid * (NPTS / 32);
  const float* pb = pts + (size_t)b * NPTS * 4;

  // -------- ordered collection: first min(count,512) in-radius indices --------
  int total = 0;
  for (int it = 0; it < NPTS / 32 / 32; ++it) {   // 16 iterations of 32 words
    const unsigned word = mw[it * 32 + lane];
    const int cnt = __popc(word);
    int scan = cnt;
#pragma unroll
    for (int d = 1; d < 32; d <<= 1) {
      const int v = __shfl_up(scan, d);
      if (lane >= d) scan += v;
    }
    const int excl = scan - cnt;
    const int wtot = __shfl(scan, 31);
    int base = total + excl;
    unsigned mm = word;
    int o = 0;
    while (mm) {
      const int j = __builtin_ctz(mm);
      mm &= mm - 1;
      const int pos = base + o;
      ++o;
      if (pos < KUP) {
        const int pidx = (it * 32 + lane) * 32 + j;
        candE[w][pos] = pb[(size_t)pidx * 4 + 3];   // energy channel
        candI[w][pos] = pidx;
      }
    }
    total += wtot;
    if (total >= KUP) break;                       // uniform across wave
  }
  const int cnt512 = total < KUP ? total : KUP;

  asm volatile("s_wait_dscnt 0x0" ::: "memory");   // cross-lane LDS handoff

  // -------- top-32 by energy (desc), tie -> lower candidate position --------
  float e[16];
  int pid[16];
#pragma unroll
  for (int k = 0; k < 16; ++k) {
    const int pos = lane + (k << 5);
    if (pos < cnt512) { e[k] = candE[w][pos]; pid[k] = candI[w][pos]; }
    else              { e[k] = neg_inf();     pid[k] = -1; }
  }
  unsigned alive = 0xffffu;
  int res = -1;                                    // lane r keeps round-r winner
  for (int r = 0; r < KS; ++r) {
    float be = neg_inf();
    int   bp = 0x7fffffff, bi2 = -1, bk = -1;
#pragma unroll
    for (int k = 0; k < 16; ++k) {
      if (((alive >> k) & 1u) && (e[k] > be)) {    // strict > keeps lower pos on ties
        be = e[k]; bp = lane + (k << 5); bi2 = pid[k]; bk = k;
      }
    }
    float we = be; int wp = bp, wi = bi2;
#pragma unroll
    for (int d = 16; d >= 1; d >>= 1) {
      const float oe = __shfl_xor(we, d);
      const int   op = __shfl_xor(wp, d);
      const int   oi = __shfl_xor(wi, d);
      if (oe > we || (oe == we && op < wp)) { we = oe; wp = op; wi = oi; }
    }
    if (lane == r) res = (we == neg_inf()) ? -1 : wi;
    if (we != neg_inf() && wp == bp) alive &= ~(1u << bk);  // owner consumes
  }

  const int plen = __popc((unsigned)__ballot(res != -1));   // = min(count,32)
  int first = __shfl(res, 0);
  int fill  = (res < 0) ? first : res;
  if (fill < 0) fill = 0;                                   // masked_gather: max(idx,0)

  if (gi < CTX) {
    float4 q = ((const float4*)pb)[fill];
    ((float4*)(out + OUT_GROUPS))[(((size_t)(b * CTX + gi)) << 5) + lane] = q;
    out[OUT_PMASK + (((size_t)(b * CTX + gi)) << 5) + lane] = (lane < plen) ? 1.0f : 0.0f;
  }
  if (lane == 0 && plen == KS) atomicAdd(&ws_glen[b], 1);
}

// ============================================================
// Kernel 4: embedding mask from full-group counts.
// ============================================================
__global__ void emb_kernel(const int* __restrict__ ws_glen, float* __restrict__ out) {
  const int t = blockIdx.x * 256 + threadIdx.x;   // 0..2047
  const int b = t >> 8, g = t & 255;
  out[OUT_EMB + t] = (g < ws_glen[b]) ? 1.0f : 0.0f;
}

// ============================================================
extern "C" void kernel_launch(void* const* d_in, const int* in_sizes, int n_in,
                              void* d_out, int out_size, void* d_ws, size_t ws_size,
                              hipStream_t stream) {
  const float* pts     = (const float*)d_in[0];
  const int*   lengths = (const int*)d_in[1];
  float* out = (float*)d_out;

  // workspace layout
  float* ws_centers        = (float*)d_ws;                                  // 32768 B
  int*   ws_glen           = (int*)((char*)d_ws + 32768);                   // 32 B
  unsigned short* ws_mask  = (unsigned short*)((char*)d_ws + 33024);        // 8 MB

  fps_kernel      <<<NB, 1024, 0, stream>>>(pts, lengths, ws_centers, ws_glen, out);
  ball_mask_kernel<<<NB * (NG / 16), 256, 0, stream>>>(pts, ws_centers, lengths, ws_mask);
  group_kernel    <<<(NB * NG) / 4, 128, 0, stream>>>(pts, (const unsigned*)ws_mask,
                                                      ws_glen, out);
  emb_kernel      <<<NB, 256, 0, stream>>>(ws_glen, out);
}